// EdgePolicy_49692771615370
// MI455X (gfx1250) — compile-verified
//
#include <hip/hip_runtime.h>
#include <hip/hip_bf16.h>
#include <math.h>

typedef float v2f __attribute__((ext_vector_type(2)));
typedef float v8f __attribute__((ext_vector_type(8)));

#define C_DIM 64

// ---------------------------------------------------------------------------
// init: deg[i] = 1.0f (reference adds +1.0 to segment_sum), agg[...] = 0
// ---------------------------------------------------------------------------
__global__ void init_kernel(float* __restrict__ deg, float* __restrict__ agg, int n) {
    int i = blockIdx.x * blockDim.x + threadIdx.x;
    int total = n * C_DIM;
    if (i < total) agg[i] = 0.0f;
    if (i < n) deg[i] = 1.0f;
}

__global__ void zero_kernel(float* __restrict__ p, int total) {
    int i = blockIdx.x * blockDim.x + threadIdx.x;
    if (i < total) p[i] = 0.0f;
}

// ---------------------------------------------------------------------------
// degree count: one thread per edge
// ---------------------------------------------------------------------------
__global__ void deg_kernel(const long long* __restrict__ ei, int nE, float* __restrict__ deg) {
    int e = blockIdx.x * blockDim.x + threadIdx.x;
    if (e >= nE) return;
    int d = (int)ei[nE + e];
    atomicAdd(&deg[d], 1.0f);
}

__global__ void dinv_kernel(float* __restrict__ deg, int n) {
    int i = blockIdx.x * blockDim.x + threadIdx.x;
    if (i < n) deg[i] = rsqrtf(deg[i]);
}

// ---------------------------------------------------------------------------
// WMMA GEMM: Y[n,64] = X[n,K] @ W[K,64], f32 via V_WMMA_F32_16X16X4_F32.
// One wave computes a 16x64 output tile (4 accumulators of 16x16).
// A 16x4 frag: lanes 0-15 -> rows, K=k..k+1; lanes 16-31 -> K=k+2..k+3.
// B 4x16 frag mirrored. C/D: VGPR r holds row r (lanes 0-15) / r+8 (16-31).
// ---------------------------------------------------------------------------
__global__ void gemm_wmma_kernel(const float* __restrict__ X, const float* __restrict__ W,
                                 float* __restrict__ Y, int n, int K) {
    const int lane = threadIdx.x & 31;
    const int wid  = threadIdx.x >> 5;
    const int tile = blockIdx.x * (blockDim.x >> 5) + wid;
    const int ntiles = n >> 4;
    if (tile >= ntiles) return;

    const int m0   = tile << 4;
    const int l    = lane & 15;
    const int half = lane >> 4;   // 0: K pair (k,k+1), 1: (k+2,k+3)

    v8f acc0 = {}; v8f acc1 = {}; v8f acc2 = {}; v8f acc3 = {};

    const float* xrow = X + (size_t)(m0 + l) * K + 2 * half;
    for (int k = 0; k < K; k += 4) {
        float2 a2 = *(const float2*)(xrow + k);
        v2f a; a.x = a2.x; a.y = a2.y;

        const float* wr0 = W + (size_t)(k + 2 * half) * C_DIM + l;
        const float* wr1 = wr0 + C_DIM;
        v2f b;
        b.x = wr0[0];  b.y = wr1[0];
        acc0 = __builtin_amdgcn_wmma_f32_16x16x4_f32(false, a, false, b, (short)0, acc0, false, false);
        b.x = wr0[16]; b.y = wr1[16];
        acc1 = __builtin_amdgcn_wmma_f32_16x16x4_f32(false, a, false, b, (short)0, acc1, false, false);
        b.x = wr0[32]; b.y = wr1[32];
        acc2 = __builtin_amdgcn_wmma_f32_16x16x4_f32(false, a, false, b, (short)0, acc2, false, false);
        b.x = wr0[48]; b.y = wr1[48];
        acc3 = __builtin_amdgcn_wmma_f32_16x16x4_f32(false, a, false, b, (short)0, acc3, false, false);
    }

    for (int r = 0; r < 8; ++r) {
        float* p = Y + (size_t)(m0 + r + 8 * half) * C_DIM + l;
        p[0]  = acc0[r];
        p[16] = acc1[r];
        p[32] = acc2[r];
        p[48] = acc3[r];
    }
}

// ---------------------------------------------------------------------------
// Edge aggregation: one wave per edge; lane handles 2 consecutive channels.
// agg[dst] += dinv[src]*dinv[dst] * xw[src]
// ---------------------------------------------------------------------------
__global__ void edge_agg_kernel(const long long* __restrict__ ei, int nE,
                                const float* __restrict__ dinv,
                                const float* __restrict__ xw,
                                float* __restrict__ agg) {
    long long tid = (long long)blockIdx.x * blockDim.x + threadIdx.x;
    int e = (int)(tid >> 5);
    if (e >= nE) return;
    int lane = threadIdx.x & 31;
    int s = (int)ei[e];
    int d = (int)ei[nE + e];
    float norm = dinv[s] * dinv[d];
    float2 v = *(const float2*)(xw + (size_t)s * C_DIM + lane * 2);
    float* ap = agg + (size_t)d * C_DIM + lane * 2;
    atomicAdd(ap,     norm * v.x);
    atomicAdd(ap + 1, norm * v.y);
}

// ---------------------------------------------------------------------------
// Node update: h = relu(agg + xw*dinv^2 + b)
// ---------------------------------------------------------------------------
__global__ void node_update_kernel(const float* __restrict__ agg, const float* __restrict__ xw,
                                   const float* __restrict__ dinv, const float* __restrict__ bias,
                                   float* __restrict__ h, int n) {
    int i = blockIdx.x * blockDim.x + threadIdx.x;
    if (i >= n * C_DIM) return;
    int nd = i >> 6;
    int c  = i & 63;
    float di = dinv[nd];
    float v = agg[i] + xw[i] * di * di + bias[c];
    h[i] = fmaxf(v, 0.0f);
}

// ---------------------------------------------------------------------------
// Ordered compaction of edges touching `node` (single block, preserves order)
// ---------------------------------------------------------------------------
__global__ void select_kernel(const long long* __restrict__ ei, int nE,
                              const int* __restrict__ node_p,
                              int* __restrict__ targets, int* __restrict__ count) {
    __shared__ int wcount[32];
    __shared__ int sBase;
    const int tid  = threadIdx.x;
    const int lane = tid & 31;
    const int wid  = tid >> 5;
    const int nw   = blockDim.x >> 5;
    const long long node = (long long)node_p[0];
    if (tid == 0) sBase = 0;
    __syncthreads();
    for (int chunk = 0; chunk < nE; chunk += blockDim.x) {
        int e = chunk + tid;
        bool match = false; int other = 0;
        if (e < nE) {
            long long s = ei[e], d = ei[nE + e];
            if (s == node)      { match = true; other = (int)d; }
            else if (d == node) { match = true; other = (int)s; }
        }
        unsigned mask = __builtin_amdgcn_ballot_w32(match);
        int lp = __popc(mask & ((1u << lane) - 1u));
        if (lane == 0) wcount[wid] = __popc(mask);
        __syncthreads();
        int woff = 0;
        for (int i = 0; i < wid; ++i) woff += wcount[i];
        if (match) targets[sBase + woff + lp] = other;
        __syncthreads();
        if (tid == 0) {
            int tot = 0;
            for (int i = 0; i < nw; ++i) tot += wcount[i];
            sBase += tot;
        }
        __syncthreads();
    }
    if (tid == 0) *count = sBase;
}

// ---------------------------------------------------------------------------
// Readout: one wave per target. score = sigmoid(h[t] . w_ro + b_ro)
// d_out[0..cap) = scores, d_out[cap..2*cap) = targets (as float)
// ---------------------------------------------------------------------------
__global__ void score_kernel(const float* __restrict__ h,
                             const float* __restrict__ w_ro, const float* __restrict__ b_ro,
                             const int* __restrict__ targets, const int* __restrict__ count,
                             float* __restrict__ out, int cap) {
    int t = blockIdx.x * (blockDim.x >> 5) + (threadIdx.x >> 5);
    int lane = threadIdx.x & 31;
    if (t >= cap || t >= *count) return;
    int tgt = targets[t];
    const float* hp = h + (size_t)tgt * C_DIM;
    float v = hp[lane] * w_ro[lane] + hp[32 + lane] * w_ro[32 + lane];
    for (int m = 16; m >= 1; m >>= 1) v += __shfl_xor(v, m, 32);
    if (lane == 0) {
        float s = 1.0f / (1.0f + __expf(-(v + b_ro[0])));
        out[t] = s;
        out[cap + t] = (float)tgt;
    }
}

// ---------------------------------------------------------------------------
extern "C" void kernel_launch(void* const* d_in, const int* in_sizes, int n_in,
                              void* d_out, int out_size, void* d_ws, size_t ws_size,
                              hipStream_t stream) {
    const float*     x    = (const float*)d_in[0];
    const long long* ei   = (const long long*)d_in[1];
    const int*       node = (const int*)d_in[3];   // value 0; low 32 bits valid either width
    const float*     w1   = (const float*)d_in[5];
    const float*     b1   = (const float*)d_in[6];
    const float*     w2   = (const float*)d_in[7];
    const float*     b2   = (const float*)d_in[8];
    const float*     w_ro = (const float*)d_in[9];
    const float*     b_ro = (const float*)d_in[10];
    float* out = (float*)d_out;

    const int n   = in_sizes[0] / 128;   // N_NODES
    const int F   = 128;                 // F_IN
    const int nE  = in_sizes[1] / 2;     // N_EDGES
    const int cap = out_size / 2;        // scores + targets concatenated

    // workspace carve-out (256B aligned)
    char* ws = (char*)d_ws;
    size_t off = 0;
    auto carve = [&](size_t bytes) -> void* {
        void* p = ws + off;
        off = (off + bytes + 255) & ~(size_t)255;
        return p;
    };
    float* deg  = (float*)carve((size_t)n * 4);           // becomes dinv in place
    float* bufA = (float*)carve((size_t)n * C_DIM * 4);   // xw1 then h1
    float* bufB = (float*)carve((size_t)n * C_DIM * 4);   // agg (both layers)
    float* bufC = (float*)carve((size_t)n * C_DIM * 4);   // xw2 then h2
    int*   tg   = (int*)carve((size_t)cap * 4);
    int*   cnt  = (int*)carve(4);

    const int nodeElems = n * C_DIM;
    const int B256 = 256;

    // degree + dinv
    init_kernel<<<(nodeElems + B256 - 1) / B256, B256, 0, stream>>>(deg, bufB, n);
    deg_kernel<<<(nE + B256 - 1) / B256, B256, 0, stream>>>(ei, nE, deg);
    dinv_kernel<<<(n + B256 - 1) / B256, B256, 0, stream>>>(deg, n);

    // layer 1: xw1 = x @ w1  (K = 128)
    {
        int ntiles = n / 16;
        int blocks = (ntiles + 3) / 4;   // 4 waves / 128-thread block
        gemm_wmma_kernel<<<blocks, 128, 0, stream>>>(x, w1, bufA, n, F);
    }
    edge_agg_kernel<<<(int)(((long long)nE * 32 + B256 - 1) / B256), B256, 0, stream>>>(ei, nE, deg, bufA, bufB);
    node_update_kernel<<<(nodeElems + B256 - 1) / B256, B256, 0, stream>>>(bufB, bufA, deg, b1, bufA, n);

    // layer 2: xw2 = h1 @ w2  (K = 64)
    zero_kernel<<<(nodeElems + B256 - 1) / B256, B256, 0, stream>>>(bufB, nodeElems);
    {
        int ntiles = n / 16;
        int blocks = (ntiles + 3) / 4;
        gemm_wmma_kernel<<<blocks, 128, 0, stream>>>(bufA, w2, bufC, n, C_DIM);
    }
    edge_agg_kernel<<<(int)(((long long)nE * 32 + B256 - 1) / B256), B256, 0, stream>>>(ei, nE, deg, bufC, bufB);
    node_update_kernel<<<(nodeElems + B256 - 1) / B256, B256, 0, stream>>>(bufB, bufC, deg, b2, bufC, n);

    // readout
    select_kernel<<<1, 1024, 0, stream>>>(ei, nE, node, tg, cnt);
    score_kernel<<<(cap + 3) / 4, 128, 0, stream>>>(bufC, w_ro, b_ro, tg, cnt, out, cap);
}